// TreeAwarePositionModule_43087111914035
// MI455X (gfx1250) — compile-verified
//
#include <hip/hip_runtime.h>
#include <math.h>

// Problem constants (from reference)
#define B_SZ     64
#define N_SZ     4096
#define D_MODEL  1024
#define H_SZ     16
#define D_HEAD   64
#define R_SZ     6
#define D_CONCAT 2048
#define NR       (N_SZ * R_SZ)          // 24576
#define LN_EPS   1e-5f
#define SCALE    0.125f                  // 1/sqrt(D_HEAD) / TEMP

typedef float v2f __attribute__((ext_vector_type(2)));
typedef float v8f __attribute__((ext_vector_type(8)));

// ---------------------------------------------------------------------------
// Kernel 1: h = LayerNorm(concat(h_t, h_vc)) * gamma + beta     [B, 2048]
// One block per batch row; 256 threads, 8 elems/thread.
// ---------------------------------------------------------------------------
__global__ void ln_kernel(const float* __restrict__ h_t,
                          const float* __restrict__ h_vc,
                          const float* __restrict__ gamma,
                          const float* __restrict__ beta,
                          float* __restrict__ h_out) {
    const int b = blockIdx.x;
    __shared__ float ws1[8], ws2[8];
    float vals[8];
    float s = 0.f, s2 = 0.f;
#pragma unroll
    for (int i = 0; i < 8; ++i) {
        int c = threadIdx.x + i * 256;
        float v = (c < D_MODEL) ? h_t[b * D_MODEL + c]
                                : h_vc[b * D_MODEL + (c - D_MODEL)];
        vals[i] = v; s += v; s2 += v * v;
    }
#pragma unroll
    for (int off = 16; off > 0; off >>= 1) {
        s  += __shfl_xor(s,  off, 32);
        s2 += __shfl_xor(s2, off, 32);
    }
    int wid = threadIdx.x >> 5, lid = threadIdx.x & 31;
    if (lid == 0) { ws1[wid] = s; ws2[wid] = s2; }
    __syncthreads();
    if (threadIdx.x == 0) {
        float a = 0.f, c2 = 0.f;
#pragma unroll
        for (int i = 0; i < 8; ++i) { a += ws1[i]; c2 += ws2[i]; }
        ws1[0] = a; ws2[0] = c2;
    }
    __syncthreads();
    const float mean = ws1[0] * (1.0f / D_CONCAT);
    const float var  = ws2[0] * (1.0f / D_CONCAT) - mean * mean;
    const float inv  = rsqrtf(var + LN_EPS);
#pragma unroll
    for (int i = 0; i < 8; ++i) {
        int c = threadIdx.x + i * 256;
        h_out[b * D_CONCAT + c] = (vals[i] - mean) * inv * gamma[c] + beta[c];
    }
}

// ---------------------------------------------------------------------------
// Kernel 2: rel_k[h,r,d] = sum_c rel_emb[r,c] * w_rel[h,c,d] + b_rel[h,d]
// grid = H*R blocks of 64 threads (lane = d, coalesced over w_rel rows).
// ---------------------------------------------------------------------------
__global__ void relk_kernel(const float* __restrict__ rel_emb,
                            const float* __restrict__ w_rel,
                            const float* __restrict__ b_rel,
                            float* __restrict__ rel_k) {
    const int h = blockIdx.x / R_SZ;
    const int r = blockIdx.x % R_SZ;
    const int d = threadIdx.x;                       // 0..63
    const float* __restrict__ w = w_rel + (size_t)h * D_MODEL * D_HEAD + d;
    const float* __restrict__ e = rel_emb + (size_t)r * D_MODEL;
    float acc = 0.f;
    for (int c = 0; c < D_MODEL; ++c)
        acc = fmaf(e[c], w[(size_t)c * D_HEAD], acc);
    rel_k[((h * R_SZ) + r) * D_HEAD + d] = acc + b_rel[h * D_HEAD + d];
}

// ---------------------------------------------------------------------------
// Kernel 3: q[b, h*64+d] = sum_c h_norm[b,c] * w_q[h,c,d] + b_q[h,d]
// fp32 WMMA: per head, GEMM [64 x 2048] x [2048 x 64].
// One wave per 16x16 output tile; V_WMMA_F32_16X16X4_F32, K-step 4.
// grid = H * 4 * 4 = 256 blocks of 32 threads (1 wave, EXEC all-ones).
// A layout (16x4 f32): lanes 0-15 M rows, VGPR0/1 = K{0,1}; lanes 16-31 K{2,3}.
// B layout (4x16 f32): lanes 0-15 N cols, VGPR0/1 = K{0,1}; lanes 16-31 K{2,3}.
// D layout: VGPR i -> (M = i + 8*half, N = lane&15).
// ---------------------------------------------------------------------------
__global__ void qproj_wmma_kernel(const float* __restrict__ h_norm,
                                  const float* __restrict__ w_q,
                                  const float* __restrict__ b_q,
                                  float* __restrict__ q) {
    const int h    = blockIdx.x >> 4;
    const int tm   = (blockIdx.x >> 2) & 3;
    const int tn   = blockIdx.x & 3;
    const int lane = threadIdx.x;
    const int half = lane >> 4;          // 0: K lo pair, 1: K hi pair
    const int l16  = lane & 15;

    const int m = tm * 16 + l16;         // A row for this lane
    const int n = tn * 16 + l16;         // B col for this lane
    const float* __restrict__ arow = h_norm + (size_t)m * D_CONCAT;
    const float* __restrict__ wcol = w_q + (size_t)h * D_CONCAT * D_HEAD + n;

    v8f acc = {};
    for (int k0 = 0; k0 < D_CONCAT; k0 += 4) {
        const int ka = k0 + half * 2;
        v2f a, bb;
        a.x  = arow[ka];
        a.y  = arow[ka + 1];
        bb.x = wcol[(size_t)ka * D_HEAD];
        bb.y = wcol[(size_t)(ka + 1) * D_HEAD];
        acc = __builtin_amdgcn_wmma_f32_16x16x4_f32(
            /*neg_a=*/false, a, /*neg_b=*/false, bb,
            /*c_mod=*/(short)0, acc, /*reuse_a=*/false, /*reuse_b=*/false);
    }
    const int nn   = tn * 16 + l16;
    const float bq = b_q[h * D_HEAD + nn];
#pragma unroll
    for (int i = 0; i < 8; ++i) {
        const int mm = tm * 16 + half * 8 + i;
        q[(size_t)mm * (H_SZ * D_HEAD) + h * D_HEAD + nn] = acc[i] + bq;
    }
}

// ---------------------------------------------------------------------------
// Kernel 4: meanRel[b,r] = (1/H) * sum_{h,d} q[b,h,d] * rel_k[h,r,d]
// grid = B*R waves of 32.
// ---------------------------------------------------------------------------
__global__ void meanrel_kernel(const float* __restrict__ q,
                               const float* __restrict__ rel_k,
                               float* __restrict__ meanRel) {
    const int b = blockIdx.x / R_SZ;
    const int r = blockIdx.x % R_SZ;
    float acc = 0.f;
    for (int idx = threadIdx.x; idx < H_SZ * D_HEAD; idx += 32) {
        const int h = idx >> 6, d = idx & 63;
        acc = fmaf(q[b * (H_SZ * D_HEAD) + idx],
                   rel_k[(h * R_SZ + r) * D_HEAD + d], acc);
    }
#pragma unroll
    for (int off = 16; off > 0; off >>= 1) acc += __shfl_xor(acc, off, 32);
    if (threadIdx.x == 0) meanRel[b * R_SZ + r] = acc * (1.0f / H_SZ);
}

// ---------------------------------------------------------------------------
// Kernel 5 (bandwidth-bound, ~1.07 GB of k_cache): 
//   meanP[b,n] = (1/H) * sum_{h,d} q[b,h,d] * k_cache[b,h,n,d]
// wave32: 32 lanes x float2 covers d=0..63 -> one fully coalesced 256B row
// per (h,n). q[b] preloaded into 16 float2 regs/lane. NPW n's per wave.
// ---------------------------------------------------------------------------
#define NPW 16
__global__ void meanp_kernel(const float* __restrict__ q,
                             const float* __restrict__ k_cache,
                             float* __restrict__ meanP) {
    const int gwave = (blockIdx.x * blockDim.x + threadIdx.x) >> 5;
    const int lane  = threadIdx.x & 31;
    const int wavesPerB = N_SZ / NPW;                    // 256
    const int b     = gwave / wavesPerB;
    const int nbase = (gwave % wavesPerB) * NPW;

    float2 qv[H_SZ];
#pragma unroll
    for (int h = 0; h < H_SZ; ++h)
        qv[h] = *(const float2*)(q + (size_t)b * (H_SZ * D_HEAD) + h * D_HEAD + 2 * lane);

    const float* __restrict__ kb =
        k_cache + (size_t)b * H_SZ * N_SZ * D_HEAD;

    for (int j = 0; j < NPW; ++j) {
        const int n = nbase + j;
        float acc = 0.f;
#pragma unroll
        for (int h = 0; h < H_SZ; ++h) {
            const float2 kv = *(const float2*)(kb + ((size_t)h * N_SZ + n) * D_HEAD + 2 * lane);
            acc = fmaf(qv[h].x, kv.x, acc);
            acc = fmaf(qv[h].y, kv.y, acc);
        }
#pragma unroll
        for (int off = 16; off > 0; off >>= 1) acc += __shfl_xor(acc, off, 32);
        if (lane == 0) meanP[b * N_SZ + n] = acc * (1.0f / H_SZ);
    }
}

// ---------------------------------------------------------------------------
// Kernel 6: logits[b,i] = (meanP[b,i/6] + meanRel[b,i%6]) * 0.125
//           probs = softmax(logits);  out = [probs | logits]
// One block per b; meanP row staged in LDS (16 KB << 320 KB/WGP).
// ---------------------------------------------------------------------------
__global__ void softmax_kernel(const float* __restrict__ meanP,
                               const float* __restrict__ meanRel,
                               float* __restrict__ out_probs,
                               float* __restrict__ out_logits) {
    const int b = blockIdx.x;
    __shared__ float sp[N_SZ];
    __shared__ float sr[R_SZ];
    __shared__ float red[8];
    for (int i = threadIdx.x; i < N_SZ; i += blockDim.x) sp[i] = meanP[b * N_SZ + i];
    if (threadIdx.x < R_SZ) sr[threadIdx.x] = meanRel[b * R_SZ + threadIdx.x];
    __syncthreads();

    const int wid = threadIdx.x >> 5, lid = threadIdx.x & 31;

    // pass 1: row max
    float m = -INFINITY;
    for (int i = threadIdx.x; i < NR; i += blockDim.x)
        m = fmaxf(m, (sp[i / R_SZ] + sr[i % R_SZ]) * SCALE);
#pragma unroll
    for (int off = 16; off > 0; off >>= 1) m = fmaxf(m, __shfl_xor(m, off, 32));
    if (lid == 0) red[wid] = m;
    __syncthreads();
    if (threadIdx.x == 0) {
        float mm = red[0];
#pragma unroll
        for (int i = 1; i < 8; ++i) mm = fmaxf(mm, red[i]);
        red[0] = mm;
    }
    __syncthreads();
    m = red[0];
    __syncthreads();

    // pass 2: sum of exp (+ write logits)
    float s = 0.f;
    for (int i = threadIdx.x; i < NR; i += blockDim.x) {
        const float v = (sp[i / R_SZ] + sr[i % R_SZ]) * SCALE;
        out_logits[(size_t)b * NR + i] = v;
        s += expf(v - m);
    }
#pragma unroll
    for (int off = 16; off > 0; off >>= 1) s += __shfl_xor(s, off, 32);
    if (lid == 0) red[wid] = s;
    __syncthreads();
    if (threadIdx.x == 0) {
        float ss = 0.f;
#pragma unroll
        for (int i = 0; i < 8; ++i) ss += red[i];
        red[0] = ss;
    }
    __syncthreads();
    const float inv = 1.0f / red[0];

    // pass 3: write probs
    for (int i = threadIdx.x; i < NR; i += blockDim.x) {
        const float v = (sp[i / R_SZ] + sr[i % R_SZ]) * SCALE;
        out_probs[(size_t)b * NR + i] = expf(v - m) * inv;
    }
}

// ---------------------------------------------------------------------------
// Launch
// ---------------------------------------------------------------------------
extern "C" void kernel_launch(void* const* d_in, const int* in_sizes, int n_in,
                              void* d_out, int out_size, void* d_ws, size_t ws_size,
                              hipStream_t stream) {
    const float* h_t      = (const float*)d_in[0];   // [B, 1024]
    const float* h_vc     = (const float*)d_in[1];   // [B, 1024]
    const float* k_cache  = (const float*)d_in[2];   // [B, H, N, 64]
    const float* rel_emb  = (const float*)d_in[3];   // [R, 1024]
    const float* w_rel    = (const float*)d_in[4];   // [H, 1024, 64]
    const float* b_rel    = (const float*)d_in[5];   // [H, 64]
    const float* w_q      = (const float*)d_in[6];   // [H, 2048, 64]
    const float* b_q      = (const float*)d_in[7];   // [H, 64]
    const float* ln_gamma = (const float*)d_in[8];   // [2048]
    const float* ln_beta  = (const float*)d_in[9];   // [2048]

    float* out_probs  = (float*)d_out;               // [B, NR]
    float* out_logits = (float*)d_out + (size_t)B_SZ * NR;

    // Workspace carve-up (floats); total ~1.82 MB
    float* ws      = (float*)d_ws;
    float* h_norm  = ws;                                   // 64*2048 = 131072
    float* q       = h_norm + (size_t)B_SZ * D_CONCAT;     // 64*1024 =  65536
    float* rel_k   = q + (size_t)B_SZ * H_SZ * D_HEAD;     // 16*6*64 =   6144
    float* meanRel = rel_k + (size_t)H_SZ * R_SZ * D_HEAD; // 64*6    =    384
    float* meanP   = meanRel + (size_t)B_SZ * R_SZ;        // 64*4096 = 262144

    // 1) LayerNorm
    ln_kernel<<<B_SZ, 256, 0, stream>>>(h_t, h_vc, ln_gamma, ln_beta, h_norm);
    // 2) relation keys
    relk_kernel<<<H_SZ * R_SZ, 64, 0, stream>>>(rel_emb, w_rel, b_rel, rel_k);
    // 3) q projection via fp32 WMMA (H * 4 * 4 tiles, 1 wave each)
    qproj_wmma_kernel<<<H_SZ * 16, 32, 0, stream>>>(h_norm, w_q, b_q, q);
    // 4) meanRel
    meanrel_kernel<<<B_SZ * R_SZ, 32, 0, stream>>>(q, rel_k, meanRel);
    // 5) meanP (HBM-bound k_cache scan): 64*256 waves, 8 waves/block
    meanp_kernel<<<(B_SZ * (N_SZ / NPW)) / 8, 256, 0, stream>>>(q, k_cache, meanP);
    // 6) logits + softmax + dual write
    softmax_kernel<<<B_SZ, 256, 0, stream>>>(meanP, meanRel, out_probs, out_logits);
}